// PointConvResBlock_33775622815928
// MI455X (gfx1250) — compile-verified
//
#include <hip/hip_runtime.h>

// Problem constants (from reference setup_inputs)
#define BB   2
#define NN   32768
#define KKN  16          // neighbors
#define CC   64          // feature channels
#define CM   32          // mid channels
#define WNC  16          // weightnet out channels
#define CIN  67          // C + 3
#define CINP 68          // padded to multiple of 4
#define MROWS (BB*NN*KKN)   // 1,048,576 (b,n,k) rows
#define NPTS  (BB*NN)       // 65,536 points
#define EPS_BN 1e-5f

typedef float v2f __attribute__((ext_vector_type(2)));
typedef float v8f __attribute__((ext_vector_type(8)));

// D = A(16x4) * B(4x16) + C, fp32 WMMA
#define WMMA_F32(a,b,c) __builtin_amdgcn_wmma_f32_16x16x4_f32(false,(a),false,(b),(short)0,(c),false,false)

// async copy: 16 bytes global -> LDS, lane-addressed; tracked by ASYNCcnt
__device__ __forceinline__ void async_b128(void* ldsDst, const void* gSrc) {
    unsigned lds = (unsigned)(unsigned long long)ldsDst;   // low 32 bits = LDS offset
    asm volatile("global_load_async_to_lds_b128 %0, %1, off"
                 :: "v"(lds), "v"(gSrc) : "memory");
}
__device__ __forceinline__ void wait_async0() {
    asm volatile("s_wait_asynccnt 0x0" ::: "memory");
}

// ---------------------------------------------------------------- k0: zero BN stat accumulators
__global__ void k0_zero(float* __restrict__ stats) {
    if (threadIdx.x < 64) stats[threadIdx.x] = 0.f;
}

// ---------------------------------------------------------------- k1: gather + geometry + z0 + MLP1/MLP2 (WMMA)
__global__ __launch_bounds__(256) void k1_geom_mlp(
    const float* __restrict__ xyz,  const float* __restrict__ feats,
    const float* __restrict__ xyzn, const int* __restrict__ nei,
    const float* __restrict__ Wm1,  const float* __restrict__ bm1,
    const float* __restrict__ Wm2,  const float* __restrict__ bm2,
    const float* __restrict__ wnW0, const float* __restrict__ wnb0,
    float* __restrict__ hbuf, float* __restrict__ zbuf, float* __restrict__ stats)
{
    __shared__ float sIn[128*CINP];   // 34816 B : [row][c]   concat(gfeat, r_miu, 0)
    __shared__ float sW1[CINP*CM];    //  8704 B : W_mlp1^T padded
    __shared__ float sH1[128*CM];     // 16384 B : relu(MLP1)
    __shared__ float sW2[CM*CM];      //  4096 B : W_mlp2^T
    __shared__ float sStat[16];
    __shared__ int   sJ[128];

    const int tid = threadIdx.x;
    const int pointBase = blockIdx.x * 8;         // block never crosses batch (N%8==0)
    const int b = pointBase >> 15;                // / NN

    for (int i = tid; i < CINP*CM; i += 256) {
        int c = i / CM, o = i % CM;
        sW1[c*CM + o] = (c < CIN) ? Wm1[o*CIN + c] : 0.f;
    }
    for (int i = tid; i < CM*CM; i += 256) {
        int c = i / CM, o = i % CM;
        sW2[c*CM + o] = Wm2[o*CM + c];
    }
    if (tid < 16) sStat[tid] = 0.f;
    if (tid < 128) {
        int p = pointBase + (tid >> 4);
        sJ[tid] = nei[(long)p*KKN + (tid & 15)];
    }
    __syncthreads();

    // gather neighbor feats (L2-resident) straight into LDS, async;
    // overlapped with the geometry VALU work below.
    for (int idx = tid; idx < 128*16; idx += 256) {
        int r = idx >> 4, q = idx & 15;
        long gj = (long)b*NN + sJ[r];
        async_b128(sIn + r*CINP + q*4, feats + gj*CC + q*4);
    }

    // geometry + weightnet layer0 pre-activation z0 (rows 0..127)
    if (tid < 128) {
        const int r = tid;
        const int p = pointBase + (r >> 4);
        const long gj = (long)b*NN + sJ[r];
        float cx = xyz[(long)p*3+0],  cy = xyz[(long)p*3+1],  cz = xyz[(long)p*3+2];
        float nx = xyzn[(long)p*3+0], ny = xyzn[(long)p*3+1], nz = xyzn[(long)p*3+2];
        float rx = xyz[gj*3+0]-cx, ry = xyz[gj*3+1]-cy, rz = xyz[gj*3+2]-cz;
        float gnx = xyzn[gj*3+0], gny = xyzn[gj*3+1], gnz = xyzn[gj*3+2];
        float rn  = sqrtf(rx*rx+ry*ry+rz*rz);
        float inv = 1.f / fmaxf(rn, 1e-12f);
        float hx = rx*inv, hy = ry*inv, hz = rz*inv;
        float dnr = nx*hx + ny*hy + nz*hz;
        float vx = nx - dnr*hx, vy = ny - dnr*hy, vz = nz - dnr*hz;
        float vi = 1.f / fmaxf(sqrtf(vx*vx+vy*vy+vz*vz), 1e-12f);
        vx *= vi; vy *= vi; vz *= vi;
        float wx = hy*vz - hz*vy, wy = hz*vx - hx*vz, wz = hx*vy - hy*vx;
        float wi = 1.f / fmaxf(sqrtf(wx*wx+wy*wy+wz*wz), 1e-12f);
        wx *= wi; wy *= wi; wz *= wi;
        float th[12];
        th[0] = gnx*nx + gny*ny + gnz*nz;
        th[1] = dnr;
        th[2] = hx*gnx + hy*gny + hz*gnz;
        th[3] = rx*nx + ry*ny + rz*nz;
        th[4] = th[2];
        th[5] = gnx*vx + gny*vy + gnz*vz;
        th[6] = gnx*wx + gny*wy + gnz*wz;
        float c8x = gny*nz - gnz*ny, c8y = gnz*nx - gnx*nz, c8z = gnx*ny - gny*nx;
        th[7] = rx*c8x + ry*c8y + rz*c8z;
        th[8] = rn;
        th[9] = rx; th[10] = ry; th[11] = rz;
        sIn[r*CINP + 64] = rx; sIn[r*CINP + 65] = ry; sIn[r*CINP + 66] = rz;
        sIn[r*CINP + 67] = 0.f;
        long rowG = (long)blockIdx.x*128 + r;
        #pragma unroll
        for (int o = 0; o < 8; ++o) {
            float acc = wnb0[o];
            #pragma unroll
            for (int c = 0; c < 12; ++c) acc += wnW0[o*12+c]*th[c];
            zbuf[rowG*16 + o] = acc;
            atomicAdd(&sStat[o], acc);
            atomicAdd(&sStat[8+o], acc*acc);
        }
    }
    wait_async0();          // gathered feats resident in LDS
    __syncthreads();
    if (tid < 16) atomicAdd(&stats[tid], sStat[tid]);

    const int lane = tid & 31, wv = tid >> 5;
    const int lo = lane & 15, hi = lane >> 4;
    const int mbase = wv * 16;

    // MLP1: [128 x 68] x [68 x 32]
    v8f c0 = {}; v8f c1 = {};
    for (int k0 = 0; k0 < CINP; k0 += 4) {
        int kk = k0 + 2*hi;
        v2f a;   a.x   = sIn[(mbase+lo)*CINP + kk];  a.y   = sIn[(mbase+lo)*CINP + kk + 1];
        v2f bb0; bb0.x = sW1[kk*CM + lo];            bb0.y = sW1[(kk+1)*CM + lo];
        v2f bb1; bb1.x = sW1[kk*CM + 16 + lo];       bb1.y = sW1[(kk+1)*CM + 16 + lo];
        c0 = WMMA_F32(a, bb0, c0);
        c1 = WMMA_F32(a, bb1, c1);
    }
    #pragma unroll
    for (int r8 = 0; r8 < 8; ++r8) {
        int row = mbase + r8 + 8*hi;
        sH1[row*CM + lo]      = fmaxf(c0[r8] + bm1[lo],    0.f);
        sH1[row*CM + 16 + lo] = fmaxf(c1[r8] + bm1[16+lo], 0.f);
    }
    __syncthreads();

    // MLP2: [128 x 32] x [32 x 32] -> hbuf
    v8f d0 = {}; v8f d1 = {};
    for (int k0 = 0; k0 < CM; k0 += 4) {
        int kk = k0 + 2*hi;
        v2f a;   a.x   = sH1[(mbase+lo)*CM + kk];  a.y   = sH1[(mbase+lo)*CM + kk + 1];
        v2f bb0; bb0.x = sW2[kk*CM + lo];          bb0.y = sW2[(kk+1)*CM + lo];
        v2f bb1; bb1.x = sW2[kk*CM + 16 + lo];     bb1.y = sW2[(kk+1)*CM + 16 + lo];
        d0 = WMMA_F32(a, bb0, d0);
        d1 = WMMA_F32(a, bb1, d1);
    }
    long rowBase = (long)blockIdx.x*128;
    #pragma unroll
    for (int r8 = 0; r8 < 8; ++r8) {
        int row = mbase + r8 + 8*hi;
        hbuf[(rowBase+row)*CM + lo]      = fmaxf(d0[r8] + bm2[lo],    0.f);
        hbuf[(rowBase+row)*CM + 16 + lo] = fmaxf(d1[r8] + bm2[16+lo], 0.f);
    }
}

// ---------------------------------------------------------------- k2: BN0 + relu + (8->8), stats for BN1
__global__ __launch_bounds__(256) void k2_bn0(
    const float* __restrict__ W, const float* __restrict__ bias,
    const float* __restrict__ g, const float* __restrict__ be,
    float* __restrict__ zbuf, float* __restrict__ stats)
{
    __shared__ float sStat[16];
    const int tid = threadIdx.x;
    if (tid < 16) sStat[tid] = 0.f;
    __syncthreads();
    long r = (long)blockIdx.x*256 + tid;
    const float invM = 1.f/(float)MROWS;
    float a0[8];
    #pragma unroll
    for (int c = 0; c < 8; ++c) {
        float m = stats[c]*invM;
        float v = stats[8+c]*invM - m*m;
        float x = zbuf[r*16 + c];
        a0[c] = fmaxf((x - m)*rsqrtf(v + EPS_BN)*g[c] + be[c], 0.f);
    }
    #pragma unroll
    for (int o = 0; o < 8; ++o) {
        float acc = bias[o];
        #pragma unroll
        for (int c = 0; c < 8; ++c) acc += W[o*8+c]*a0[c];
        zbuf[r*16 + o] = acc;
        atomicAdd(&sStat[o], acc);
        atomicAdd(&sStat[8+o], acc*acc);
    }
    __syncthreads();
    if (tid < 16) atomicAdd(&stats[16+tid], sStat[tid]);
}

// ---------------------------------------------------------------- k3: BN1 + relu + (8->16), stats for BN2
__global__ __launch_bounds__(256) void k3_bn1(
    const float* __restrict__ W, const float* __restrict__ bias,
    const float* __restrict__ g, const float* __restrict__ be,
    float* __restrict__ zbuf, float* __restrict__ stats)
{
    __shared__ float sStat[32];
    const int tid = threadIdx.x;
    if (tid < 32) sStat[tid] = 0.f;
    __syncthreads();
    long r = (long)blockIdx.x*256 + tid;
    const float invM = 1.f/(float)MROWS;
    float a1[8];
    #pragma unroll
    for (int c = 0; c < 8; ++c) {
        float m = stats[16+c]*invM;
        float v = stats[24+c]*invM - m*m;
        float x = zbuf[r*16 + c];
        a1[c] = fmaxf((x - m)*rsqrtf(v + EPS_BN)*g[c] + be[c], 0.f);
    }
    float z2[16];
    #pragma unroll
    for (int o = 0; o < 16; ++o) {
        float acc = bias[o];
        #pragma unroll
        for (int c = 0; c < 8; ++c) acc += W[o*8+c]*a1[c];
        z2[o] = acc;
    }
    #pragma unroll
    for (int o = 0; o < 16; ++o) {
        zbuf[r*16 + o] = z2[o];
        atomicAdd(&sStat[o], z2[o]);
        atomicAdd(&sStat[16+o], z2[o]*z2[o]);
    }
    __syncthreads();
    if (tid < 32) atomicAdd(&stats[32+tid], sStat[tid]);
}

// ---------------------------------------------------------------- k4: BN2 -> w, agg = h^T w (WMMA), final linear (WMMA) + relu + residual
__global__ __launch_bounds__(256) void k4_agg_out(
    const float* __restrict__ feats, const float* __restrict__ g, const float* __restrict__ be,
    const float* __restrict__ Wlin, const float* __restrict__ blin,
    const float* __restrict__ hbuf, const float* __restrict__ zbuf,
    const float* __restrict__ stats, float* __restrict__ out)
{
    // phase A: sMem[0:8192) = h tile [16pt][16k][32cm]; sMem[8192:12288) = w tile [16pt][16k][16]
    // phase B: sMem[0:8192) reused as agg [16pt][512]
    __shared__ float sMem[12288];   // 48 KB
    const int tid = threadIdx.x;
    const int pointBase = blockIdx.x * 16;
    long rowG = (long)pointBase*KKN + tid;   // 256 (b,n,k) rows

    // kick off the h tile (largest HBM stream) as async global->LDS
    {
        const float* gsrc = hbuf + rowG*32;
        float* ldst = sMem + tid*32;
        #pragma unroll
        for (int q = 0; q < 8; ++q)
            async_b128(ldst + q*4, gsrc + q*4);
    }

    // warm L2/L1 with W_lin (reused by every k-step of the final GEMM)
    for (int i = tid; i < 1024; i += 256)
        __builtin_prefetch(Wlin + i*32, 0, 3);

    // w = relu(BN2(z2)) while the async copy is in flight
    {
        const float invM = 1.f/(float)MROWS;
        #pragma unroll
        for (int c = 0; c < 16; ++c) {
            float m = stats[32+c]*invM;
            float v = stats[48+c]*invM - m*m;
            float x = zbuf[rowG*16 + c];
            sMem[8192 + tid*16 + c] = fmaxf((x - m)*rsqrtf(v + EPS_BN)*g[c] + be[c], 0.f);
        }
    }
    wait_async0();
    __syncthreads();

    const int lane = tid & 31, wv = tid >> 5;
    const int lo = lane & 15, hi = lane >> 4;

    // agg per point: (32cm x 16w) = h^T(16k x 32cm) * w(16k x 16); 32 tile-jobs over 8 waves
    v8f acc[4] = {{},{},{},{}};
    for (int jj = 0; jj < 4; ++jj) {
        int j = wv*4 + jj;
        int p = j >> 1, t = j & 1;
        v8f c = {};
        for (int k0 = 0; k0 < 16; k0 += 4) {
            int kk = k0 + 2*hi;
            v2f a;  a.x  = sMem[p*512 + kk*32     + t*16 + lo];
                    a.y  = sMem[p*512 + (kk+1)*32 + t*16 + lo];
            v2f bb; bb.x = sMem[8192 + p*256 + kk*16     + lo];
                    bb.y = sMem[8192 + p*256 + (kk+1)*16 + lo];
            c = WMMA_F32(a, bb, c);
        }
        acc[jj] = c;
    }
    __syncthreads();                 // all reads of h/w done
    for (int jj = 0; jj < 4; ++jj) {
        int j = wv*4 + jj;
        int p = j >> 1, t = j & 1;
        #pragma unroll
        for (int r8 = 0; r8 < 8; ++r8) {
            int m = t*16 + r8 + 8*hi;              // cm index
            sMem[p*512 + m*16 + lo] = acc[jj][r8]; // agg[p][cm*16 + wch]
        }
    }
    __syncthreads();

    // final: out(16pt x 64) = agg(16 x 512) * Wlin^T(512 x 64); waves 0..3, one 16-col tile each
    if (wv < 4) {
        const int nt = wv;
        v8f c = {};
        for (int k0 = 0; k0 < 512; k0 += 4) {
            int kk = k0 + 2*hi;
            v2f a;  a.x  = sMem[lo*512 + kk];  a.y = sMem[lo*512 + kk + 1];
            v2f bb; bb.x = Wlin[(long)(nt*16 + lo)*512 + kk];
                    bb.y = Wlin[(long)(nt*16 + lo)*512 + kk + 1];
            c = WMMA_F32(a, bb, c);
        }
        #pragma unroll
        for (int r8 = 0; r8 < 8; ++r8) {
            int m = r8 + 8*hi;                  // point within tile
            int o = nt*16 + lo;
            long gp = pointBase + m;
            out[gp*CC + o] = fmaxf(c[r8] + blin[o], 0.f) + feats[gp*CC + o];
        }
    }
}

// ---------------------------------------------------------------- host launch
extern "C" void kernel_launch(void* const* d_in, const int* in_sizes, int n_in,
                              void* d_out, int out_size, void* d_ws, size_t ws_size,
                              hipStream_t stream)
{
    const float* xyz   = (const float*)d_in[0];
    const float* feats = (const float*)d_in[1];
    const float* xyzn  = (const float*)d_in[2];
    const int*   nei   = (const int*)d_in[3];
    const float* Wm1   = (const float*)d_in[4];
    const float* bm1   = (const float*)d_in[5];
    const float* Wm2   = (const float*)d_in[6];
    const float* bm2   = (const float*)d_in[7];
    const float* wnW0  = (const float*)d_in[8];
    const float* wnb0  = (const float*)d_in[9];
    const float* wng0  = (const float*)d_in[10];
    const float* wnbe0 = (const float*)d_in[11];
    const float* wnW1  = (const float*)d_in[12];
    const float* wnb1  = (const float*)d_in[13];
    const float* wng1  = (const float*)d_in[14];
    const float* wnbe1 = (const float*)d_in[15];
    const float* wnW2  = (const float*)d_in[16];
    const float* wnb2  = (const float*)d_in[17];
    const float* wng2  = (const float*)d_in[18];
    const float* wnbe2 = (const float*)d_in[19];
    const float* Wlin  = (const float*)d_in[20];
    const float* blin  = (const float*)d_in[21];
    float* out = (float*)d_out;

    float* stats = (float*)d_ws;                  // 64 floats of BN accumulators
    float* hbuf  = stats + 256;                   // MROWS x 32
    float* zbuf  = hbuf + (size_t)MROWS*32;       // MROWS x 16

    k0_zero   <<<1, 64, 0, stream>>>(stats);
    k1_geom_mlp<<<NPTS/8,    256, 0, stream>>>(xyz, feats, xyzn, nei, Wm1, bm1, Wm2, bm2,
                                               wnW0, wnb0, hbuf, zbuf, stats);
    k2_bn0    <<<MROWS/256, 256, 0, stream>>>(wnW1, wnb1, wng0, wnbe0, zbuf, stats);
    k3_bn1    <<<MROWS/256, 256, 0, stream>>>(wnW2, wnb2, wng1, wnbe1, zbuf, stats);
    k4_agg_out<<<NPTS/16,   256, 0, stream>>>(feats, wng2, wnbe2, Wlin, blin,
                                              hbuf, zbuf, stats, out);
}